// SPUTransformer_14654428414206
// MI455X (gfx1250) — compile-verified
//
#include <hip/hip_runtime.h>

// ---------------------------------------------------------------------------
// SPU bound-relaxation transformer (elementwise, bandwidth-bound).
// Input : bounds (N,2) f32 interleaved [l,u]  ->  64 MB
// Output: (N,6) f32 [out_lo,out_hi,slope_lo,slope_hi,shift_lo,shift_hi] -> 192 MB
// HBM roofline @23.3 TB/s ~= 11 us; compute is negligible -> pure streaming.
// Strategy: 2 elements/thread => 1x global_load_b128 + 3x global_store_b128
// per lane, all non-temporal (single-pass stream, keep 192MB L2 clean).
// Note: use clang ext_vector float4 (v4f) — HIP's float4 class is rejected by
// __builtin_nontemporal_load/store.
// ---------------------------------------------------------------------------

typedef float v4f __attribute__((ext_vector_type(4)));

__device__ __forceinline__ void spu_relax(float l, float u, float o[6]) {
    // spu(x)      = x>=0 ? x*x - 0.5 : sigmoid(-x) - 1
    // der_spu(x)  = x>=0 ? 2x        : -s*(1-s), s = sigmoid(-x) = 1/(1+e^x)
    const float sl = 1.0f / (1.0f + __expf(l));
    const float su = 1.0f / (1.0f + __expf(u));
    const bool lpos = (l >= 0.0f);
    const bool upos = (u >= 0.0f);
    float vl = lpos ? fmaf(l, l, -0.5f) : (sl - 1.0f);
    float vu = upos ? fmaf(u, u, -0.5f) : (su - 1.0f);
    const float tl = lpos ? (2.0f * l) : (-sl * (1.0f - sl));
    const float tu = upos ? (2.0f * u) : (-su * (1.0f - su));

    const float diff = u - l;
    const bool neg   = (u < 0.0f);
    const bool pos   = lpos;
    const bool cross = !(neg || pos);

    const float all_slopes = (vu - vl) / diff;
    const bool cross_pos = (all_slopes > 0.0f) && cross;
    const bool cross_neg = (all_slopes < 0.0f) && cross;

    float slope_lo = pos ? tl : 0.0f;
    float slope_hi = pos ? all_slopes : 0.0f;
    slope_lo = neg ? all_slopes : slope_lo;
    slope_hi = neg ? tl : slope_hi;

    const bool cnc  = (all_slopes <  tl) && cross_neg;
    const bool cnns = (all_slopes >= tl) && cross_neg;
    const bool cpns = cross_pos || cnns;

    slope_hi = cpns ? all_slopes : slope_hi;
    const float shifts_temp = cpns ? fmaf(-slope_hi, u, vu) : 0.0f;
    const float safe_sh = cpns ? slope_hi : 1.0f;
    float ix = cpns ? ((-shifts_temp - 0.5f) / safe_sh) : 0.0f;

    float var1   = cross_pos ? (0.5f * fabsf(vu + 0.5f) * (u - ix)) : 0.0f;
    float cutoff = cross_pos ? (0.5f * fabsf(vl + 0.5f) * (l - ix)) : 0.0f;
    var1   = cnns ? (0.5f * fabsf(vl + 0.5f) * (ix - l)) : var1;
    cutoff = cnns ? (0.5f * fabsf(vu + 0.5f) * (ix - u)) : cutoff;
    var1   = cpns ? (var1 - cutoff) : var1;

    float yit = cpns ? fmaf(-tu, u, vu) : 0.0f;
    const float new_yl = cpns ? fmaf(tu, l, yit) : 0.0f;
    const float var2 = cpns ? (0.5f * fabsf(new_yl - vl) * diff) : 0.0f;
    const bool ind_v2 = cpns && (var1 < var2);
    const bool ind_v1 = cpns && (var2 <= var1);
    slope_lo = ind_v2 ? 0.0f : slope_lo;
    slope_lo = ind_v1 ? tu   : slope_lo;

    slope_hi = cnc ? tl : slope_hi;
    const float area_box = cnc ? (fabsf(vl + 0.5f) * diff) : 0.0f;
    yit = cnc ? fmaf(-tl, l, vl) : yit;
    const float safe_tl = cnc ? tl : 1.0f;
    ix = cnc ? ((-yit - 0.5f) / safe_tl) : ix;
    var1   = cnc ? (0.5f * fabsf(vl + 0.5f) * (ix - l)) : var1;
    cutoff = cnc ? (0.5f * fabsf(vu + 0.5f) * (ix - u)) : cutoff;
    var1   = cnc ? (var1 - cutoff) : var1;

    const bool tri_smaller = cnc && (var1 < area_box);
    const float sqrt_arg = tri_smaller ? fmaf(tl, tl, 4.0f * (yit + 0.5f)) : 1.0f;
    const float new_bound =
        tri_smaller ? (0.5f * (tl + sqrtf(fmaxf(sqrt_arg, 0.0f)))) : 0.0f;
    // spu(new_bound) (selected only under tri_smaller)
    const float snb = 1.0f / (1.0f + __expf(new_bound));
    const float vnb = (new_bound >= 0.0f) ? fmaf(new_bound, new_bound, -0.5f)
                                          : (snb - 1.0f);
    u  = tri_smaller ? new_bound : u;
    vu = tri_smaller ? vnb       : vu;

    const bool box_leq = cnc && (area_box <= var1);
    slope_lo = box_leq ? 0.0f : slope_lo;
    slope_hi = box_leq ? 0.0f : slope_hi;

    const bool switched = (all_slopes < 0.0f);   // original all_slopes
    const float vl2 = switched ? vu : vl;
    const float vu2 = switched ? vl : vu;
    const float l2  = switched ? u  : l;
    const float u2  = switched ? l  : u;

    const float out_lo = cross ? -0.5f : vl2;
    const float out_hi = vu2;
    const float shift_hi = fmaf(-slope_hi, u2, vu2);
    float shift_lo = fmaf(-slope_lo, l2, vl2);
    shift_lo = cross_neg ? -0.5f : shift_lo;
    shift_lo = ind_v2 ? -0.5f : shift_lo;
    shift_lo = ind_v1 ? yit   : shift_lo;

    o[0] = out_lo;  o[1] = out_hi;
    o[2] = slope_lo; o[3] = slope_hi;
    o[4] = shift_lo; o[5] = shift_hi;
}

__global__ __launch_bounds__(256) void spu_transformer_kernel(
    const float* __restrict__ bounds, float* __restrict__ out, int n) {
    const int t = blockIdx.x * blockDim.x + threadIdx.x;
    const int base = t * 2;  // element index
    if (base + 1 < n) {
        // One b128 NT load: two (l,u) pairs.
        const v4f p = __builtin_nontemporal_load(
            reinterpret_cast<const v4f*>(bounds) + t);
        float o0[6], o1[6];
        spu_relax(p.x, p.y, o0);
        spu_relax(p.z, p.w, o1);
        // 12 contiguous output floats -> three b128 NT stores (48B-aligned).
        v4f* ov = reinterpret_cast<v4f*>(out + (size_t)base * 6u);
        v4f s0; s0.x = o0[0]; s0.y = o0[1]; s0.z = o0[2]; s0.w = o0[3];
        v4f s1; s1.x = o0[4]; s1.y = o0[5]; s1.z = o1[0]; s1.w = o1[1];
        v4f s2; s2.x = o1[2]; s2.y = o1[3]; s2.z = o1[4]; s2.w = o1[5];
        __builtin_nontemporal_store(s0, ov + 0);
        __builtin_nontemporal_store(s1, ov + 1);
        __builtin_nontemporal_store(s2, ov + 2);
    } else if (base < n) {
        // Tail (N odd — not hit for N = 8388608, kept for generality).
        const float l = bounds[(size_t)base * 2u + 0u];
        const float u = bounds[(size_t)base * 2u + 1u];
        float o0[6];
        spu_relax(l, u, o0);
#pragma unroll
        for (int k = 0; k < 6; ++k) out[(size_t)base * 6u + k] = o0[k];
    }
}

extern "C" void kernel_launch(void* const* d_in, const int* in_sizes, int n_in,
                              void* d_out, int out_size, void* d_ws, size_t ws_size,
                              hipStream_t stream) {
    (void)n_in; (void)out_size; (void)d_ws; (void)ws_size;
    const float* bounds = (const float*)d_in[0];
    float* out = (float*)d_out;
    const int n = in_sizes[0] / 2;          // number of (l,u) intervals
    const int threads_needed = (n + 1) / 2; // 2 elements per thread
    const int block = 256;                  // 8 wave32 waves per block
    const int grid = (threads_needed + block - 1) / block;
    spu_transformer_kernel<<<grid, block, 0, stream>>>(bounds, out, n);
}